// LowRankDynamicKernelOnFeatures_44444321579365
// MI455X (gfx1250) — compile-verified
//
#include <hip/hip_runtime.h>
#include <math.h>

typedef float v2f __attribute__((ext_vector_type(2)));
typedef float v8f __attribute__((ext_vector_type(8)));

#define NN 16384
#define DD 64
#define RR 32
#define TT 32
#define BB 8
#define OO 12
#define HID 128
#define EPSF 1e-6f

// Branchless softplus on raw hw trans ops (v_exp_f32/v_log_f32 are base-2).
// log2 argument is in [1,2] -> no denormal guard needed, unlike __logf.
__device__ __forceinline__ float softplus_f(float x) {
    const float t = __builtin_amdgcn_exp2f(-fabsf(x) * 1.442695040888963f);
    return fmaxf(x, 0.f) + 0.6931471805599453f * __builtin_amdgcn_logf(1.f + t);
}

// ---------------------------------------------------------------------------
// Phase 1: per (b, chunk of 512 rows): partialU[b][chunk][32*64] = Bmat_chunk^T @ H_chunk.
// b==0 blocks also materialize Bmat = softplus(basis) (2 MB, stays in L2 for phase 3)
// and Bsum column-sum partials.
// 8 waves; each wave owns 64 K-rows and the full 32x64 output (2 r-tiles x 4 d-tiles).
// WMMA f32 16x16x4: A[m=r][k=n], B[k=n][d].
// ---------------------------------------------------------------------------
__global__ __launch_bounds__(256) void k_phase1(const float* __restrict__ H,
                                                const float* __restrict__ basis,
                                                float* __restrict__ pU,
                                                float* __restrict__ pBS,
                                                float* __restrict__ Bm) {
    __shared__ float red[4][2048];     // 32 KB (two-round reduce)
    __shared__ float bsred[8][2][32];  // 2 KB

    const int blk = blockIdx.x;
    const int b = blk >> 5;
    const int chunk = blk & 31;
    const int tid = threadIdx.x;
    const int w = tid >> 5;
    const int lane = tid & 31;
    const int half = lane >> 4;   // 0: K={0,1}, 1: K={2,3}
    const int Lm = lane & 15;
    const int klo = half * 2;
    const int nwave = chunk * 512 + w * 64;
    const bool writeBmat = (b == 0);

    const v8f vzero = {0.f, 0.f, 0.f, 0.f, 0.f, 0.f, 0.f, 0.f};
    v8f acc[2][4];
#pragma unroll
    for (int rt = 0; rt < 2; ++rt)
#pragma unroll
        for (int dt = 0; dt < 4; ++dt) acc[rt][dt] = vzero;

    float bs0 = 0.f, bs1 = 0.f;
    const float* __restrict__ Hb = H + (size_t)b * NN * DD;

    for (int s = 0; s < 16; ++s) {
        const int n0 = nwave + 4 * s;
        const size_t r0a = (size_t)(n0 + klo) * RR;
        const size_t r0b = (size_t)(n0 + klo + 1) * RR;
        // A fragments (softplus(basis)); each chunk element touched once per wave
        const float a00 = softplus_f(basis[r0a + Lm]);
        const float a01 = softplus_f(basis[r0b + Lm]);
        const float a10 = softplus_f(basis[r0a + 16 + Lm]);
        const float a11 = softplus_f(basis[r0b + 16 + Lm]);
        if (writeBmat) {
            Bm[r0a + Lm] = a00;
            Bm[r0b + Lm] = a01;
            Bm[r0a + 16 + Lm] = a10;
            Bm[r0b + 16 + Lm] = a11;
        }
        bs0 += a00 + a01;
        bs1 += a10 + a11;
        const v2f A0 = {a00, a01};
        const v2f A1 = {a10, a11};
#pragma unroll
        for (int dt = 0; dt < 4; ++dt) {
            v2f Bf;
            Bf.x = Hb[(size_t)(n0 + klo) * DD + dt * 16 + Lm];
            Bf.y = Hb[(size_t)(n0 + klo + 1) * DD + dt * 16 + Lm];
            acc[0][dt] = __builtin_amdgcn_wmma_f32_16x16x4_f32(
                false, A0, false, Bf, (short)0, acc[0][dt], false, false);
            acc[1][dt] = __builtin_amdgcn_wmma_f32_16x16x4_f32(
                false, A1, false, Bf, (short)0, acc[1][dt], false, false);
        }
    }

    bsred[w][0][lane] = bs0;
    bsred[w][1][lane] = bs1;

    // Two-round cross-wave reduction (keeps static LDS at 34 KB).
    if (w < 4) {
#pragma unroll
        for (int rt = 0; rt < 2; ++rt)
#pragma unroll
            for (int dt = 0; dt < 4; ++dt)
#pragma unroll
                for (int j = 0; j < 8; ++j) {
                    const int r = rt * 16 + j + half * 8;
                    const int d = dt * 16 + Lm;
                    red[w][r * 64 + d] = acc[rt][dt][j];
                }
    }
    __syncthreads();
    float part[8];
#pragma unroll
    for (int k = 0; k < 8; ++k) {
        const int i = tid + 256 * k;
        part[k] = red[0][i] + red[1][i] + red[2][i] + red[3][i];
    }
    __syncthreads();
    if (w >= 4) {
#pragma unroll
        for (int rt = 0; rt < 2; ++rt)
#pragma unroll
            for (int dt = 0; dt < 4; ++dt)
#pragma unroll
                for (int j = 0; j < 8; ++j) {
                    const int r = rt * 16 + j + half * 8;
                    const int d = dt * 16 + Lm;
                    red[w - 4][r * 64 + d] = acc[rt][dt][j];
                }
    }
    __syncthreads();
    float* __restrict__ outp = pU + ((size_t)b * 32 + chunk) * 2048;
#pragma unroll
    for (int k = 0; k < 8; ++k) {
        const int i = tid + 256 * k;
        outp[i] = part[k] + red[0][i] + red[1][i] + red[2][i] + red[3][i];
    }
    if (b == 0 && tid < 32) {
        float v = 0.f;
        if (tid < 16) {
            for (int ww = 0; ww < 8; ++ww) v += bsred[ww][0][tid] + bsred[ww][0][tid + 16];
        } else {
            for (int ww = 0; ww < 8; ++ww) v += bsred[ww][1][tid - 16] + bsred[ww][1][tid];
        }
        pBS[chunk * 32 + tid] = v;
    }
}

// ---------------------------------------------------------------------------
// Phase 2 (tiny, one WG per batch): reduce partials -> U, ctx, MLP, write
// SU = s (*) U  and  w = s * Bsum.
// ---------------------------------------------------------------------------
__global__ __launch_bounds__(256) void k_phase2(const float* __restrict__ ts_out,
                                                const float* __restrict__ W1,
                                                const float* __restrict__ b1,
                                                const float* __restrict__ W2,
                                                const float* __restrict__ b2,
                                                const float* __restrict__ pU,
                                                const float* __restrict__ pBS,
                                                float* __restrict__ U,
                                                float* __restrict__ Wdeg,
                                                float* __restrict__ SU) {
    __shared__ float sU[2048];
    __shared__ float sBsum[32];
    __shared__ float sCtx[32];
    __shared__ float sFeat[OO][64];
    __shared__ float sH[OO][HID];
    __shared__ float sS[OO][32];

    const int b = blockIdx.x;
    const int tid = threadIdx.x;

    for (int i = tid; i < 2048; i += 256) {
        float v = 0.f;
        const float* p = pU + (size_t)b * 32 * 2048 + i;
#pragma unroll 4
        for (int c = 0; c < 32; ++c) v += p[(size_t)c * 2048];
        sU[i] = v;
        U[(size_t)b * 2048 + i] = v;
    }
    if (tid < 32) {
        float v = 0.f;
        for (int c = 0; c < 32; ++c) v += pBS[c * 32 + tid];
        sBsum[tid] = v;
    }
    __syncthreads();
    if (tid < 32) {
        float v = 0.f;
        for (int d = 0; d < 64; ++d) {
            const float u = sU[tid * 64 + d];
            v += u * u;
        }
        sCtx[tid] = sqrtf(v * (1.f / 64.f) + EPSF);
    }
    __syncthreads();
    for (int i = tid; i < OO * 64; i += 256) {
        const int o = i >> 6, c = i & 63;
        sFeat[o][c] = (c < TT) ? ts_out[((size_t)b * OO + o) * TT + c] : sCtx[c - TT];
    }
    __syncthreads();
    for (int i = tid; i < OO * HID; i += 256) {
        const int o = i >> 7, j = i & 127;
        float v = b1[j];
        for (int k = 0; k < 64; ++k) v += sFeat[o][k] * W1[k * HID + j];
        sH[o][j] = 0.5f * v * (1.f + erff(v * 0.70710678118654752f));  // exact gelu
    }
    __syncthreads();
    for (int i = tid; i < OO * 32; i += 256) {
        const int o = i >> 5, r = i & 31;
        float v = b2[r];
        for (int k = 0; k < HID; ++k) v += sH[o][k] * W2[k * 32 + r];
        const float s = softplus_f(v) + EPSF;
        sS[o][r] = s;
        Wdeg[((size_t)b * OO + o) * 32 + r] = s * sBsum[r];
    }
    __syncthreads();
    for (int i = tid; i < OO * 2048; i += 256) {
        const int o = i / 2048, rd = i & 2047, r = rd >> 6;
        SU[(size_t)b * OO * 2048 + i] = sS[o][r] * sU[rd];
    }
}

// ---------------------------------------------------------------------------
// Phase 3 (the bandwidth phase): per (b, o, 512-row chunk):
//   M = Bmat_tile @ SU ; deg on-the-fly from A-fragment loads; blend with H.
// SU B-fragments and deg weights are loop-invariant registers. A-fragments are
// single b64 loads from the precomputed Bmat (L2-resident). Output stored with
// non-temporal hint (write-once stream, keep L2 for H/Bmat reuse).
// ---------------------------------------------------------------------------
__global__ __launch_bounds__(256) void k_phase3(const float* __restrict__ H,
                                                const float* __restrict__ Bm,
                                                const float* __restrict__ SU,
                                                const float* __restrict__ Wdeg,
                                                const float* __restrict__ alphap,
                                                float* __restrict__ out) {
    const int blk = blockIdx.x;
    const int b = blk / (OO * 32);      // b outermost -> H[b] stays hot in L2 across o
    const int o = (blk / 32) % OO;
    const int chunk = blk & 31;
    const int tid = threadIdx.x;
    const int w = tid >> 5;
    const int lane = tid & 31;
    const int half = lane >> 4;
    const int Lm = lane & 15;
    const int klo = half * 2;

    const float alpha = fminf(fmaxf(alphap[0], 0.f), 1.f);
    const float* __restrict__ SUp = SU + ((size_t)b * OO + o) * 2048;

    // deg weights: lane r holds w[r]; resolve loop-invariant per-K scalars once
    const float wv = Wdeg[((size_t)b * OO + o) * 32 + lane];
    float wA[8], wB[8];
#pragma unroll
    for (int kk = 0; kk < 8; ++kk) {
        wA[kk] = __shfl(wv, 4 * kk + klo, 32);
        wB[kk] = __shfl(wv, 4 * kk + klo + 1, 32);
    }

    // Loop-invariant B fragments (SU), 64 VGPRs
    v2f Bf[4][8];
#pragma unroll
    for (int dt = 0; dt < 4; ++dt)
#pragma unroll
        for (int kk = 0; kk < 8; ++kk) {
            const int r0 = 4 * kk + klo;
            Bf[dt][kk].x = SUp[r0 * 64 + dt * 16 + Lm];
            Bf[dt][kk].y = SUp[(r0 + 1) * 64 + dt * 16 + Lm];
        }

    const float* __restrict__ Hb = H + (size_t)b * NN * DD;
    float* __restrict__ outp = out + ((size_t)b * OO + o) * (size_t)NN * DD;
    const v8f vzero = {0.f, 0.f, 0.f, 0.f, 0.f, 0.f, 0.f, 0.f};

    for (int it = 0; it < 4; ++it) {
        const int n0 = chunk * 512 + it * 128 + w * 16;
        v8f acc[4];
#pragma unroll
        for (int dt = 0; dt < 4; ++dt) acc[dt] = vzero;
        float dacc = 0.f;

        const float* __restrict__ arow = Bm + (size_t)(n0 + Lm) * RR + klo;
#pragma unroll
        for (int kk = 0; kk < 8; ++kk) {
            // A fragment: one aligned b64 load of {Bmat[row, 4kk+klo], Bmat[row, 4kk+klo+1]}
            const v2f Af = *(const v2f*)(arow + 4 * kk);
            dacc += wA[kk] * Af.x + wB[kk] * Af.y;
#pragma unroll
            for (int dt = 0; dt < 4; ++dt)
                acc[dt] = __builtin_amdgcn_wmma_f32_16x16x4_f32(
                    false, Af, false, Bf[dt][kk], (short)0, acc[dt], false, false);
        }

        // lane m and m+16 each hold half of deg[row n0+m]
        const float deg = dacc + __shfl_xor(dacc, 16, 32);

#pragma unroll
        for (int j = 0; j < 8; ++j) {
            const int jrow = j + half * 8;
            const float dj = __shfl(deg, jrow, 32);  // lane jrow holds deg[row n0+jrow]
            const float inv = 1.f / (dj + EPSF);
            const int row = n0 + jrow;
#pragma unroll
            for (int dt = 0; dt < 4; ++dt) {
                const int col = dt * 16 + Lm;
                const float hval = Hb[(size_t)row * DD + col];
                const float m = acc[dt][j] * inv;
                __builtin_nontemporal_store(hval + alpha * (m - hval),
                                            outp + (size_t)row * DD + col);
            }
        }
    }
}

// ---------------------------------------------------------------------------
extern "C" void kernel_launch(void* const* d_in, const int* in_sizes, int n_in,
                              void* d_out, int out_size, void* d_ws, size_t ws_size,
                              hipStream_t stream) {
    const float* H      = (const float*)d_in[0];
    const float* ts_out = (const float*)d_in[1];
    const float* basis  = (const float*)d_in[2];
    const float* W1     = (const float*)d_in[3];
    const float* b1     = (const float*)d_in[4];
    const float* W2     = (const float*)d_in[5];
    const float* b2     = (const float*)d_in[6];
    const float* alpha  = (const float*)d_in[7];
    float* out = (float*)d_out;
    float* ws = (float*)d_ws;

    // workspace layout (floats), ~4.8 MB total; every region written before read
    float* pU   = ws;                     // 8*32*2048   = 524288
    float* pBS  = ws + 524288;            // 32*32       = 1024
    float* Bmat = ws + 525312;            // 16384*32    = 524288
    float* U    = ws + 1049600;           // 8*2048      = 16384
    float* Wdeg = ws + 1065984;           // 8*12*32     = 3072
    float* SU   = ws + 1069056;           // 8*12*2048   = 196608

    k_phase1<<<dim3(BB * 32), dim3(256), 0, stream>>>(H, basis, pU, pBS, Bmat);
    k_phase2<<<dim3(BB), dim3(256), 0, stream>>>(ts_out, W1, b1, W2, b2, pU, pBS, U, Wdeg, SU);
    k_phase3<<<dim3(BB * OO * 32), dim3(256), 0, stream>>>(H, Bmat, SU, Wdeg, alpha, out);
}